// VectorQuantizer_81449759802152
// MI455X (gfx1250) — compile-verified
//
#include <hip/hip_runtime.h>

typedef float v2f __attribute__((ext_vector_type(2)));
typedef float v8f __attribute__((ext_vector_type(8)));

#define VOCAB      512
#define DIM        192
#define K2         96          // DIM/2 : float2 elements per row
#define CHUNK      64          // codes staged in LDS per pass
#define NCHUNK     8           // VOCAB / CHUNK
#define TILES      4           // 16-code WMMA column tiles per chunk
#define LDS_STRIDE 80          // float2 stride per k2-row (bank padding: +16)
#define ROWS_BLK   128         // rows per block (8 waves * 16)

// ---------------------------------------------------------------------------
// Kernel 1: ehalf[v] = 0.5 * sum_k codebook[v][k]^2
// ---------------------------------------------------------------------------
__global__ __launch_bounds__(256) void vq_norms_kernel(
    const float* __restrict__ cb, float* __restrict__ ehalf) {
  int v = blockIdx.x * blockDim.x + threadIdx.x;
  if (v >= VOCAB) return;
  const float4* row = (const float4*)(cb + (size_t)v * DIM);
  float s = 0.0f;
#pragma unroll
  for (int i = 0; i < DIM / 4; ++i) {
    float4 t = row[i];
    s += t.x * t.x + t.y * t.y + t.z * t.z + t.w * t.w;
  }
  ehalf[v] = 0.5f * s;
}

// ---------------------------------------------------------------------------
// Kernel 2: WMMA fp32 GEMM (scores = z . e^T - 0.5|e|^2) + running argmax,
// then gather z_q and emit indices.
// ---------------------------------------------------------------------------
__global__ __launch_bounds__(256) void vq_main_kernel(
    const float* __restrict__ z, const float* __restrict__ cb,
    const float* __restrict__ ehalf, float* __restrict__ out_zq,
    float* __restrict__ out_idx) {
  // K-major staged codebook chunk: [k2 in 0..95][code in 0..63], padded rows
  __shared__ v2f ldsB[K2 * LDS_STRIDE];  // 60 KB

  const int tid  = threadIdx.x;
  const int lane = tid & 31;
  const int wave = tid >> 5;
  const int l16  = lane & 15;   // row (A) / column (B,D) within tile
  const int kh   = lane >> 4;   // which K half-pair this lane holds

  const int rowW = blockIdx.x * ROWS_BLK + wave * 16;  // first row of this wave

  // --- Preload A panel: 16 rows x 192 K into 48 fp32 A-fragments (regs) ---
  // A frag k holds K = 4k + 2*kh + {0,1} for row rowW + l16.
  v2f aF[48];
  {
    const v2f* zr = (const v2f*)z + (size_t)(rowW + l16) * K2;
#pragma unroll
    for (int k = 0; k < 48; ++k) aF[k] = zr[2 * k + kh];
  }

  float bv[8];
  int   bi[8];
#pragma unroll
  for (int i = 0; i < 8; ++i) { bv[i] = -3.4e38f; bi[i] = 0; }

  const v2f* cb2 = (const v2f*)cb;

  for (int ch = 0; ch < NCHUNK; ++ch) {
    __syncthreads();  // protect previous chunk's readers
    // Stage 64 codes, K-major, consecutive tids -> consecutive codes (no LDS
    // write conflicts). Exact trip count 24 for every thread (no divergence).
    for (int e = tid; e < CHUNK * K2; e += 256) {
      int c  = e & (CHUNK - 1);
      int k2 = e >> 6;
      ldsB[k2 * LDS_STRIDE + c] = cb2[(size_t)(ch * CHUNK + c) * K2 + k2];
    }
    __syncthreads();

#pragma unroll
    for (int t = 0; t < TILES; ++t) {
      v8f acc = {};
#pragma unroll
      for (int k = 0; k < 48; ++k) {
        // B frag: lane holds codes t*16+l16 at K = 4k + 2*kh + {0,1}
        v2f b = ldsB[(2 * k + kh) * LDS_STRIDE + t * 16 + l16];
        acc = __builtin_amdgcn_wmma_f32_16x16x4_f32(
            false, aF[k], false, b, (short)0, acc, false, false);
      }
      const int col = ch * CHUNK + t * 16 + l16;   // candidate code id
      const float eh = ehalf[col];
#pragma unroll
      for (int i = 0; i < 8; ++i) {
        float s = acc[i] - eh;  // rows i (lanes 0-15) / 8+i (lanes 16-31)
        if (s > bv[i]) { bv[i] = s; bi[i] = col; }
      }
    }
  }

  // --- Cross-lane argmax within each 16-lane half (rows 0-7 | 8-15) ---
#pragma unroll
  for (int i = 0; i < 8; ++i) {
    float v = bv[i];
    int   ix = bi[i];
#pragma unroll
    for (int m = 8; m >= 1; m >>= 1) {
      float ov = __shfl_xor(v, m, 32);
      int   oi = __shfl_xor(ix, m, 32);
      if (ov > v || (ov == v && oi < ix)) { v = ov; ix = oi; }
    }
    bv[i] = v;
    bi[i] = ix;
  }

  // Broadcast winning index for each of the wave's 16 rows to all lanes.
  int idxRow[16];
#pragma unroll
  for (int i = 0; i < 8; ++i) {
    idxRow[i]     = __shfl(bi[i], 0, 32);   // rows 0..7  live in lanes 0-15
    idxRow[8 + i] = __shfl(bi[i], 16, 32);  // rows 8..15 live in lanes 16-31
  }

  if (lane == 0) {
#pragma unroll
    for (int r = 0; r < 16; ++r)
      out_idx[rowW + r] = (float)idxRow[r];
  }

  // --- Gather z_q rows (coalesced float2 copies, 3 iters/lane/row) ---
  v2f* oz = (v2f*)out_zq;
#pragma unroll
  for (int r = 0; r < 16; ++r) {
    const v2f* src = cb2 + (size_t)idxRow[r] * K2;
    v2f* dst = oz + (size_t)(rowW + r) * K2;
    for (int e = lane; e < K2; e += 32) dst[e] = src[e];
  }
}

// ---------------------------------------------------------------------------
extern "C" void kernel_launch(void* const* d_in, const int* in_sizes, int n_in,
                              void* d_out, int out_size, void* d_ws,
                              size_t ws_size, hipStream_t stream) {
  const float* z  = (const float*)d_in[0];  // (64,32,32,192) fp32
  const float* cb = (const float*)d_in[1];  // (512,192) fp32

  float* ehalf   = (float*)d_ws;            // 512 floats of scratch
  float* out_zq  = (float*)d_out;           // N*192 floats
  float* out_idx = out_zq + (size_t)65536 * DIM;  // then N indices

  vq_norms_kernel<<<2, 256, 0, stream>>>(cb, ehalf);
  vq_main_kernel<<<512, 256, 0, stream>>>(z, cb, ehalf, out_zq, out_idx);
}